// BiasedFeatureDropout_42296837931097
// MI455X (gfx1250) — compile-verified
//
#include <hip/hip_runtime.h>
#include <stdint.h>

// BiasedFeatureDropout on [64,256,56,56] f32 — gfx1250.
// Memory-bound streaming op: 411 MB traffic -> ~17.6us floor @ 23.3 TB/s.
// Inline JAX threefry2x32 (key=(0,1)) is independent of loaded data, so the
// ~1000 VALU ops per thread hide the 8 in-flight B128 NT loads completely.
// NT hints everywhere: tensor (205MB) > L2 (192MB), zero reuse.
//
// Layout tricks (all exact):
//  * JAX pairs counters (i, i+N/2); N/2 = 32 full batches => pair partners
//    share a channel.
//  * kHalf/4 = 2048*3136 with 2048 % 256 == 0 => chunks j + q*kHalf/4 for
//    q=0..3 ALL share one channel => one threshold covers all 32 elements
//    a thread touches.
//  * u < p collapses to one u32 compare on the raw threefry bits.
//  * x * (keep ? 1.25f : 0.0f) reproduces x*mask*scale bit-for-bit (incl -0.0).

typedef float v4f __attribute__((ext_vector_type(4)));

static constexpr uint32_t kHW      = 56u * 56u;          // 3136 (mult of 4)
static constexpr uint32_t kC       = 256u;
static constexpr uint32_t kN       = 64u * kC * kHW;     // 51,380,224
static constexpr uint32_t kHalf    = kN / 2u;            // 25,690,112
static constexpr uint32_t kQuarter = kHalf / 4u;         // 6,422,528 = 2048*3136

// u < 0.2f  <=>  (bits>>9) < 0.2f*2^23 = 1677721.625  <=>  bits < 1677722<<9
static constexpr uint32_t kThrBias = 858993664u;
// u < 0.8f  <=>  (bits>>9) < 6710886.5                <=>  bits < 6710887<<9
static constexpr uint32_t kThrReg  = 3435974144u;

__device__ __forceinline__ uint32_t rotl32(uint32_t v, uint32_t r) {
  return (v << r) | (v >> (32u - r));   // v_alignbit_b32
}

// JAX threefry2x32, key = (0,1): ks = [0, 1, 0^1^0x1BD11BDA].
__device__ __forceinline__ void threefry2x32_k01(uint32_t c0, uint32_t c1,
                                                 uint32_t& o0, uint32_t& o1) {
  constexpr uint32_t ks0 = 0u, ks1 = 1u, ks2 = 0x1BD11BDBu;
  uint32_t x0 = c0 + ks0;
  uint32_t x1 = c1 + ks1;
  auto rnd = [&](uint32_t r) { x0 += x1; x1 = rotl32(x1, r); x1 ^= x0; };
  rnd(13); rnd(15); rnd(26); rnd(6);
  x0 += ks1; x1 += ks2 + 1u;
  rnd(17); rnd(29); rnd(16); rnd(24);
  x0 += ks2; x1 += ks0 + 2u;
  rnd(13); rnd(15); rnd(26); rnd(6);
  x0 += ks0; x1 += ks1 + 3u;
  rnd(17); rnd(29); rnd(16); rnd(24);
  x0 += ks1; x1 += ks2 + 4u;
  rnd(13); rnd(15); rnd(26); rnd(6);
  x0 += ks2; x1 += ks0 + 5u;
  o0 = x0; o1 = x1;
}

// One thread = 32 elements: float4 at j + q*kQuarter and its pair partner at
// +kHalf, for q = 0..3. Lane-consecutive float4s => perfectly coalesced.
__global__ __launch_bounds__(256, 1)
void biased_dropout_threefry_kernel(const float* __restrict__ x,
                                    float* __restrict__ y) {
  const uint32_t j = (blockIdx.x * 256u + threadIdx.x) * 4u;
  if (j >= kQuarter) return;

  // Issue all 8 streaming loads up front (one clause, 8 B128 NT in flight);
  // the threefry math below is independent and hides their latency.
  v4f a[4], b[4];
#pragma unroll
  for (uint32_t q = 0; q < 4u; ++q) {
    a[q] = __builtin_nontemporal_load((const v4f*)(x + j + q * kQuarter));
    b[q] = __builtin_nontemporal_load((const v4f*)(x + j + q * kQuarter + kHalf));
  }

  // One threshold for all 32 elements (see header comment).
  const uint32_t ch  = (j / kHW) & (kC - 1u);
  const uint32_t thr = (ch < 32u) ? kThrBias : kThrReg;

#pragma unroll
  for (uint32_t q = 0; q < 4u; ++q) {
    const uint32_t base = j + q * kQuarter;
    uint32_t m0[4], m1[4];
#pragma unroll
    for (uint32_t k = 0; k < 4u; ++k)
      threefry2x32_k01(base + k, base + k + kHalf, m0[k], m1[k]);

    v4f ra, rb;
#pragma unroll
    for (uint32_t k = 0; k < 4u; ++k) {
      ra[k] = a[q][k] * (m0[k] < thr ? 1.25f : 0.0f);
      rb[k] = b[q][k] * (m1[k] < thr ? 1.25f : 0.0f);
    }
    __builtin_nontemporal_store(ra, (v4f*)(y + base));
    __builtin_nontemporal_store(rb, (v4f*)(y + base + kHalf));
  }
}

extern "C" void kernel_launch(void* const* d_in, const int* in_sizes, int n_in,
                              void* d_out, int out_size, void* d_ws, size_t ws_size,
                              hipStream_t stream) {
  (void)in_sizes; (void)n_in; (void)d_ws; (void)ws_size; (void)out_size;
  const float* x = (const float*)d_in[0];
  float*       y = (float*)d_out;

  const uint32_t threads = kQuarter / 4u;            // 1,605,632 (exact)
  const uint32_t blocks  = (threads + 255u) / 256u;  // 6,272, no tail
  biased_dropout_threefry_kernel<<<blocks, 256, 0, stream>>>(x, y);
}